// MixtureModelDecoder_74053826118259
// MI455X (gfx1250) — compile-verified
//
#include <hip/hip_runtime.h>
#include <hip/hip_bf16.h>
#include <math.h>
#include <stdint.h>

typedef float v2f __attribute__((ext_vector_type(2)));
typedef float v8f __attribute__((ext_vector_type(8)));
typedef unsigned int v4u __attribute__((ext_vector_type(4)));
typedef int v4i __attribute__((ext_vector_type(4)));
typedef int v8i __attribute__((ext_vector_type(8)));

#define B_     128
#define ETA_   64
#define KCLS_  32
#define HID_   256
#define NATOM_ 24
#define NBOND_ 5
#define NA_    5120    // B*N
#define NE_    99840   // B*N*(N-1)/2
#define EPS_   1e-5f

// ---------------------------------------------------------------------------
// V_WMMA_F32_16X16X4_F32 wrapper.
// A: 16x4 (M = lane%16; lanes<16 hold K={k0,k0+1}, lanes>=16 hold K={k0+2,k0+3})
// B: 4x16 (N = lane%16; VGPR v holds row k0+v (lanes<16) / k0+v+2 (lanes>=16))
// C/D: 16x16, VGPR r = row r (lanes<16) / row r+8 (lanes>=16), col = lane%16.
// ---------------------------------------------------------------------------
__device__ __forceinline__ v8f wmma4(v2f a, v2f b, v8f c) {
  return __builtin_amdgcn_wmma_f32_16x16x4_f32(false, a, false, b, (short)0, c,
                                               false, false);
}

// ---------------------------------------------------------------------------
// Tensor Data Mover: 1D (single-row tile) DMA of `ndwords` dwords into LDS.
// Builds a D# per CDNA5 ISA: group0 {count=1, lds_addr,
// global_addr[56:0], type=2}; group1 {data_size=4B, tensor_dim0=ndwords,
// tensor_dim1=1, tile_dim0=ndwords, tile_dim1=1, stride0=ndwords}.
// Tracked with TENSORcnt; caller must s_wait_tensorcnt + barrier.
// ---------------------------------------------------------------------------
__device__ __forceinline__ void tdm_load_1d(float* lds_dst, const float* gsrc,
                                            unsigned int ndwords) {
  unsigned long long ga = (unsigned long long)(uintptr_t)gsrc;
  unsigned int laddr = (unsigned int)(uintptr_t)lds_dst;  // addr[31:0] = LDS byte offset
  v4u g0;
  g0.x = 1u;                                              // count=1 (valid)
  g0.y = laddr;                                           // lds_addr
  g0.z = (unsigned int)ga;                                // global_addr[31:0]
  g0.w = ((unsigned int)(ga >> 32) & 0x01FFFFFFu) | (2u << 30);  // [56:32]|type=2
  v8i g1;
  g1[0] = (int)(2u << 16);                                // data_size = 4 bytes
  g1[1] = (int)((ndwords & 0xFFFFu) << 16);               // tensor_dim0[15:0]
  g1[2] = (int)(((ndwords >> 16) & 0xFFFFu) | (1u << 16));// tensor_dim0[31:16], tensor_dim1=1
  g1[3] = (int)((ndwords & 0xFFFFu) << 16);               // tile_dim0
  g1[4] = 1;                                              // tile_dim1 = 1
  g1[5] = (int)ndwords;                                   // tensor_dim0_stride[31:0]
  g1[6] = 0;
  g1[7] = 0;
  v4i zz = {0, 0, 0, 0};
#if __clang_major__ >= 23
  v8i zz8 = {0, 0, 0, 0, 0, 0, 0, 0};
  __builtin_amdgcn_tensor_load_to_lds(g0, g1, zz, zz, zz8, 0);
#else
  __builtin_amdgcn_tensor_load_to_lds(g0, g1, zz, zz, 0);
#endif
}

// First-layer GEMM tile: (16 rows x K=32) @ (32 x HID) -> one 16-wide n-tile.
// Xt: per-wave LDS tile, row-major 16x32.
// Wp: LDS, pair-interleaved [k>>1][n][k&1] (stride 512 per k-pair).
__device__ __forceinline__ v8f gemm_k32p(const float* Xt, const float* Wp,
                                         int n0, int lo, int hi) {
  v8f acc = {0.f, 0.f, 0.f, 0.f, 0.f, 0.f, 0.f, 0.f};
#pragma unroll
  for (int k0 = 0; k0 < 32; k0 += 4) {
    v2f a = *(const v2f*)&Xt[lo * 32 + k0 + 2 * hi];
    v2f b = *(const v2f*)&Wp[((k0 + 2 * hi) >> 1) * 512 + (n0 + lo) * 2];
    acc = wmma4(a, b, acc);
  }
  return acc;
}

// ---------------------------------------------------------------------------
// Kernel 1: class MLP + log_softmax (128 rows; scalar, single workgroup)
// ---------------------------------------------------------------------------
__global__ void mmd_class_mlp(const float* eta, const float* cw1,
                              const float* cb1, const float* cg,
                              const float* cbe, const float* ca,
                              const float* cw2, const float* cb2,
                              float* log_pi) {
  extern __shared__ float lds[];
  float* sH = lds;           // 256*128 col-major [c][r]
  float* sLg = lds + 32768;  // 128*32
  const int t = threadIdx.x;
  {
    float w[ETA_];
#pragma unroll
    for (int k = 0; k < ETA_; ++k) w[k] = cw1[k * HID_ + t];
    float sum = 0.f, sq = 0.f;
    for (int r = 0; r < B_; ++r) {
      float acc = cb1[t];
#pragma unroll
      for (int k = 0; k < ETA_; ++k) acc += eta[r * ETA_ + k] * w[k];
      sH[t * B_ + r] = acc;
      sum += acc;
      sq += acc * acc;
    }
    float m = sum / (float)B_;
    float var = sq / (float)B_ - m * m;
    float rstd = rsqrtf(var + EPS_);
    float gg = cg[t], bb = cbe[t], al = *ca;
    for (int r = 0; r < B_; ++r) {
      float v = (sH[t * B_ + r] - m) * rstd * gg + bb;
      sH[t * B_ + r] = v >= 0.f ? v : al * v;
    }
  }
  __syncthreads();
  for (int i = t; i < B_ * KCLS_; i += 256) {
    int r = i >> 5, j = i & 31;
    float acc = cb2[j];
    for (int c = 0; c < HID_; ++c) acc += sH[c * B_ + r] * cw2[c * KCLS_ + j];
    sLg[i] = acc;
  }
  __syncthreads();
  if (t < B_) {
    float m = -3.0e38f;
    for (int j = 0; j < KCLS_; ++j) m = fmaxf(m, sLg[t * KCLS_ + j]);
    float s = 0.f;
    for (int j = 0; j < KCLS_; ++j) s += expf(sLg[t * KCLS_ + j] - m);
    float lse = m + logf(s);
    for (int j = 0; j < KCLS_; ++j)
      log_pi[t * KCLS_ + j] = sLg[t * KCLS_ + j] - lse;
  }
}

// ---------------------------------------------------------------------------
// Kernel 2: gumbel straight-through + reparameterized z. One wave per node.
// ---------------------------------------------------------------------------
__global__ void mmd_latent(const float* log_pi, const float* gumbel_u,
                           const float* noise, const int* ngi, const float* cm,
                           const float* cls, float* z) {
  const int wave = threadIdx.x >> 5, lane = threadIdx.x & 31;
  const int a = blockIdx.x * 8 + wave;
  const int gidx = ngi[a];
  const float u = gumbel_u[a * 32 + lane];
  const float l = log_pi[gidx * 32 + lane] + (-logf(-logf(u)));  // TAU==1
  float m = l;
#pragma unroll
  for (int off = 16; off; off >>= 1) m = fmaxf(m, __shfl_xor(m, off));
  const float e = expf(l - m);
  float s = e;
#pragma unroll
  for (int off = 16; off; off >>= 1) s += __shfl_xor(s, off);
  const float y = e / s;
  float bv = l;
  int bi = lane;
#pragma unroll
  for (int off = 16; off; off >>= 1) {
    float ov = __shfl_xor(bv, off);
    int oi = __shfl_xor(bi, off);
    if (ov > bv || (ov == bv && oi < bi)) { bv = ov; bi = oi; }
  }
  const float ys = __shfl(y, bi);
  const float carg = (1.0f - ys) + ys;
  const float mu = carg * cm[bi * 32 + lane];
  float lsg = carg * cls[bi * 32 + lane];
  lsg = fminf(fmaxf(lsg, -20.f), 30.f);
  z[a * 32 + lane] = noise[a * 32 + lane] * expf(lsg) + mu;
}

// ---------------------------------------------------------------------------
// Kernel 3: symmetrize + pair-interleave: out[m*64 + d*2 + (ab&1)] =
//           0.5*(bm[d,a,b]+bm[d,b,a]) with ab = 2m+delta, a=ab>>5, b=ab&31.
// ---------------------------------------------------------------------------
__global__ void mmd_wsym(const float* bm, float* wflatp) {
  int i = blockIdx.x * 256 + threadIdx.x;  // < 32768
  int m = i >> 6, r = i & 63, d = r >> 1, del = r & 1;
  int ab = 2 * m + del, a2 = ab >> 5, b2 = ab & 31;
  wflatp[i] =
      0.5f * (bm[d * 1024 + a2 * 32 + b2] + bm[d * 1024 + b2 * 32 + a2]);
}

// ---------------------------------------------------------------------------
// Kernel 3b: pair-interleave MLP weights. W1 (32x256) -> W1p; W2 (256xNout,
// zero-padded to 32 cols) -> W2p. 8192 threads-worth each.
// ---------------------------------------------------------------------------
__global__ void mmd_repack(const float* W1, float* W1p, const float* W2,
                           int Nout, float* W2p) {
  int i = blockIdx.x * 256 + threadIdx.x;  // < 8192
  {
    int k = i >> 8, n = i & 255;
    W1p[(k >> 1) * 512 + n * 2 + (k & 1)] = W1[i];
  }
  {
    int k = i >> 5, n = i & 31;
    W2p[(k >> 1) * 64 + n * 2 + (k & 1)] = (n < Nout) ? W2[k * Nout + n] : 0.f;
  }
}

// ---------------------------------------------------------------------------
// Kernel 4: edge bilinear as (E x 1024) @ (1024 x 32) WMMA GEMM.
// Wflatp staged via TDM (128 KB); A rows = zi⊗zj outer products on the fly.
// ---------------------------------------------------------------------------
__global__ void mmd_bilinear(const float* z, const int* eidx,
                             const float* wflatp, float* efeat) {
  extern __shared__ float lds[];
  float* sW = lds;          // 32768, pair-interleaved [m][d][delta]
  float* sZ = lds + 32768;  // 8 waves * (zi 16x32 + zj 16x32)
  const int wave = threadIdx.x >> 5, lane = threadIdx.x & 31;
  const int lo = lane & 15, hi = lane >> 4;
  const int e0 = (blockIdx.x * 8 + wave) * 16;
  if (wave == 0) tdm_load_1d(sW, wflatp, 32768);
  float* zi = sZ + wave * 1024;
  float* zj = zi + 512;
  for (int i = lane; i < 512; i += 32) {
    int e = i >> 5, d = i & 31;
    zi[i] = z[eidx[(e0 + e) * 2 + 0] * 32 + d];
    zj[i] = z[eidx[(e0 + e) * 2 + 1] * 32 + d];
  }
  __builtin_amdgcn_s_wait_tensorcnt(0);
  __syncthreads();
  v8f acc0 = {0.f, 0.f, 0.f, 0.f, 0.f, 0.f, 0.f, 0.f};
  v8f acc1 = {0.f, 0.f, 0.f, 0.f, 0.f, 0.f, 0.f, 0.f};
  for (int a = 0; a < 32; ++a) {
    const float zia = zi[lo * 32 + a];
#pragma unroll
    for (int b0 = 0; b0 < 32; b0 += 4) {
      const int m = (a * 32 + b0 + 2 * hi) >> 1;
      v2f zp = *(const v2f*)&zj[lo * 32 + b0 + 2 * hi];
      v2f af;
      af.x = zia * zp.x;
      af.y = zia * zp.y;
      v2f bf0 = *(const v2f*)&sW[m * 64 + lo * 2];
      v2f bf1 = *(const v2f*)&sW[m * 64 + (16 + lo) * 2];
      acc0 = wmma4(af, bf0, acc0);
      acc1 = wmma4(af, bf1, acc1);
    }
  }
#pragma unroll
  for (int r = 0; r < 8; ++r) {
    efeat[(e0 + r + 8 * hi) * 32 + lo] = acc0[r];
    efeat[(e0 + r + 8 * hi) * 32 + 16 + lo] = acc1[r];
  }
}

// ---------------------------------------------------------------------------
// Kernel 5: BN statistics: column sum/sumsq of H = X@W1 + b1 (WMMA).
// W1p via TDM (block), X tiles via per-wave TDM. 8 waves/block.
// ---------------------------------------------------------------------------
__global__ void mmd_mlp_stats(const float* X, int M, const float* W1p,
                              const float* b1, float* stats) {
  extern __shared__ float lds[];
  float* sW1 = lds;        // 8192 pair-interleaved
  float* sX = lds + 8192;  // 8 * 512
  const int wave = threadIdx.x >> 5, lane = threadIdx.x & 31;
  const int lo = lane & 15, hi = lane >> 4;
  const int row0 = (blockIdx.x * 8 + wave) * 16;
  float* Xt = sX + wave * 512;
  if (wave == 0) tdm_load_1d(sW1, W1p, 8192);
  tdm_load_1d(Xt, X + (size_t)row0 * 32, 512);
  __builtin_amdgcn_s_wait_tensorcnt(0);
  __syncthreads();
  for (int n0 = 0; n0 < HID_; n0 += 16) {
    v8f h = gemm_k32p(Xt, sW1, n0, lo, hi);
    const float bias = b1[n0 + lo];
    float s = 0.f, sq = 0.f;
#pragma unroll
    for (int r = 0; r < 8; ++r) {
      float v = h[r] + bias;
      s += v;
      sq += v * v;
    }
    s += __shfl_xor(s, 16);
    sq += __shfl_xor(sq, 16);
    if (lane < 16) {
      atomicAdd(&stats[n0 + lo], s);
      atomicAdd(&stats[HID_ + n0 + lo], sq);
    }
  }
}

// ---------------------------------------------------------------------------
// Kernel 6: finish: recompute H (WMMA), BN+PReLU, LDS round-trip, second
// GEMM (WMMA, N zero-padded), optional row softmax. 4 waves/block.
// ---------------------------------------------------------------------------
__global__ void mmd_mlp_finish(const float* X, int M, const float* W1p,
                               const float* b1, const float* g,
                               const float* be, const float* alpha,
                               const float* W2p, const float* b2,
                               const float* stats, float* out, int Nout,
                               int do_softmax) {
  extern __shared__ float lds[];
  float* sW1 = lds;          // 8192 pair-interleaved
  float* sW2 = lds + 8192;   // 8192 pair-interleaved (padded to 32 cols)
  float* sX = lds + 16384;   // 4 * 512
  float* sHb = lds + 18432;  // 4 * 16*260
  const int wave = threadIdx.x >> 5, lane = threadIdx.x & 31;
  const int lo = lane & 15, hi = lane >> 4;
  const int row0 = (blockIdx.x * 4 + wave) * 16;
  float* Xt = sX + wave * 512;
  float* Ht = sHb + wave * 16 * 260;
  if (wave == 0) {
    tdm_load_1d(sW1, W1p, 8192);
    tdm_load_1d(sW2, W2p, 8192);
  }
  tdm_load_1d(Xt, X + (size_t)row0 * 32, 512);
  __builtin_amdgcn_s_wait_tensorcnt(0);
  __syncthreads();
  const float Mf = (float)M;
  const float al = *alpha;
  for (int n0 = 0; n0 < HID_; n0 += 16) {
    v8f h = gemm_k32p(Xt, sW1, n0, lo, hi);
    const int n = n0 + lo;
    const float bias = b1[n];
    const float mean = stats[n] / Mf;
    const float var = stats[HID_ + n] / Mf - mean * mean;
    const float rstd = rsqrtf(var + EPS_);
    const float gg = g[n], bb = be[n];
#pragma unroll
    for (int r = 0; r < 8; ++r) {
      float v = (h[r] + bias - mean) * rstd * gg + bb;
      v = v >= 0.f ? v : al * v;
      Ht[(r + 8 * hi) * 260 + n] = v;
    }
  }
  const int NT2 = (Nout + 15) >> 4;
  for (int nt = 0; nt < NT2; ++nt) {
    v8f acc = {0.f, 0.f, 0.f, 0.f, 0.f, 0.f, 0.f, 0.f};
#pragma unroll 8
    for (int k0 = 0; k0 < HID_; k0 += 4) {
      v2f a = *(const v2f*)&Ht[lo * 260 + k0 + 2 * hi];
      v2f b = *(const v2f*)&sW2[((k0 + 2 * hi) >> 1) * 64 + (nt * 16 + lo) * 2];
      acc = wmma4(a, b, acc);
    }
    const int col = nt * 16 + lo;
    if (!do_softmax) {
      if (col < Nout) {
#pragma unroll
        for (int r = 0; r < 8; ++r)
          out[(size_t)(row0 + r + 8 * hi) * Nout + col] = acc[r] + b2[col];
      }
    } else {
#pragma unroll
      for (int r = 0; r < 8; ++r) {
        float v = (col < Nout) ? acc[r] + b2[col] : -3.0e38f;
        float mx = v;
#pragma unroll
        for (int off = 1; off < 16; off <<= 1)
          mx = fmaxf(mx, __shfl_xor(mx, off));
        float ev = expf(v - mx);
        float sm = ev;
#pragma unroll
        for (int off = 1; off < 16; off <<= 1) sm += __shfl_xor(sm, off);
        if (col < Nout)
          out[(size_t)(row0 + r + 8 * hi) * Nout + col] = ev / sm;
      }
    }
  }
}

// ---------------------------------------------------------------------------
// launch
// ---------------------------------------------------------------------------
extern "C" void kernel_launch(void* const* d_in, const int* in_sizes, int n_in,
                              void* d_out, int out_size, void* d_ws,
                              size_t ws_size, hipStream_t stream) {
  (void)in_sizes; (void)n_in; (void)out_size; (void)ws_size;
  const float* eta = (const float*)d_in[0];
  const float* gumbel_u = (const float*)d_in[1];
  const float* noise_z = (const float*)d_in[2];
  const int* edge_index = (const int*)d_in[3];
  const int* node_graph_idx = (const int*)d_in[4];
  const float* cw1 = (const float*)d_in[5];
  const float* cb1 = (const float*)d_in[6];
  const float* cg = (const float*)d_in[7];
  const float* cbe = (const float*)d_in[8];
  const float* ca = (const float*)d_in[9];
  const float* cw2 = (const float*)d_in[10];
  const float* cb2 = (const float*)d_in[11];
  const float* cm = (const float*)d_in[12];
  const float* cls = (const float*)d_in[13];
  const float* aw1 = (const float*)d_in[14];
  const float* ab1 = (const float*)d_in[15];
  const float* ag = (const float*)d_in[16];
  const float* abe = (const float*)d_in[17];
  const float* aa = (const float*)d_in[18];
  const float* aw2 = (const float*)d_in[19];
  const float* ab2 = (const float*)d_in[20];
  const float* bm = (const float*)d_in[21];
  const float* bw1 = (const float*)d_in[22];
  const float* bb1 = (const float*)d_in[23];
  const float* bg = (const float*)d_in[24];
  const float* bbe = (const float*)d_in[25];
  const float* ba = (const float*)d_in[26];
  const float* bw2 = (const float*)d_in[27];
  const float* bb2 = (const float*)d_in[28];

  float* ws = (float*)d_ws;
  float* ws_logpi = ws;            // 4096
  float* ws_z = ws + 4096;         // 163840
  float* ws_wflatp = ws + 167936;  // 32768
  float* ws_efeat = ws + 200704;   // 3194880
  float* ws_stats = ws + 3395584;  // 512 atom + 512 bond
  float* ws_stats_a = ws_stats;
  float* ws_stats_b = ws_stats + 512;
  float* ws_aw1p = ws + 3396608;  // 8192
  float* ws_aw2p = ws + 3404800;  // 8192
  float* ws_bw1p = ws + 3412992;  // 8192
  float* ws_bw2p = ws + 3421184;  // 8192

  float* out_atom = (float*)d_out;                 // 5120*24
  float* out_bond = (float*)d_out + NA_ * NATOM_;  // 99840*5

  hipMemsetAsync(ws_stats, 0, 1024 * sizeof(float), stream);

  mmd_class_mlp<<<1, 256, (32768 + 4096) * sizeof(float), stream>>>(
      eta, cw1, cb1, cg, cbe, ca, cw2, cb2, ws_logpi);
  mmd_latent<<<NA_ / 8, 256, 0, stream>>>(ws_logpi, gumbel_u, noise_z,
                                          node_graph_idx, cm, cls, ws_z);
  mmd_wsym<<<128, 256, 0, stream>>>(bm, ws_wflatp);
  mmd_repack<<<32, 256, 0, stream>>>(aw1, ws_aw1p, aw2, NATOM_, ws_aw2p);
  mmd_repack<<<32, 256, 0, stream>>>(bw1, ws_bw1p, bw2, NBOND_, ws_bw2p);

  mmd_bilinear<<<NE_ / (8 * 16), 256, (32768 + 8192) * sizeof(float), stream>>>(
      ws_z, edge_index, ws_wflatp, ws_efeat);

  const size_t stats_lds = (8192 + 4096) * sizeof(float);
  mmd_mlp_stats<<<NA_ / (8 * 16), 256, stats_lds, stream>>>(
      ws_z, NA_, ws_aw1p, ab1, ws_stats_a);
  mmd_mlp_stats<<<NE_ / (8 * 16), 256, stats_lds, stream>>>(
      ws_efeat, NE_, ws_bw1p, bb1, ws_stats_b);

  const size_t fin_lds = (8192 + 8192 + 2048 + 4 * 16 * 260) * sizeof(float);
  mmd_mlp_finish<<<NA_ / (4 * 16), 128, fin_lds, stream>>>(
      ws_z, NA_, ws_aw1p, ab1, ag, abe, aa, ws_aw2p, ab2, ws_stats_a, out_atom,
      NATOM_, 0);
  mmd_mlp_finish<<<NE_ / (4 * 16), 128, fin_lds, stream>>>(
      ws_efeat, NE_, ws_bw1p, bb1, bg, bbe, ba, ws_bw2p, bb2, ws_stats_b,
      out_bond, NBOND_, 1);
}